// JSD_63823214019076
// MI455X (gfx1250) — compile-verified
//
#include <hip/hip_runtime.h>

// Problem constants (match reference: H=W=2048, sigma=1 -> 1/(2*sigma^2)=0.5)
#define IMG_H 2048
#define IMG_W 2048
#define TILE  64
#define HS    (TILE + 2)   // halo-tile stride (66)

typedef float v2f __attribute__((ext_vector_type(2)));
typedef float v8f __attribute__((ext_vector_type(8)));

// -0.5 * log2(e): exp(-d2/2) == exp2(C2*d2); log2(p) == C2*d2 - log2(S)
#define C2  (-0.72134752044448170367996234050095f)
#define LN2 ( 0.69314718055994530941723212145818f)

// Native base-2 hardware transcendentals (no scale-multiply overhead).
__device__ __forceinline__ float hw_exp2(float x) { return __builtin_amdgcn_exp2f(x); } // v_exp_f32
__device__ __forceinline__ float hw_log2(float x) { return __builtin_amdgcn_logf(x); }  // v_log_f32

// Stage-1: per-block JS divergence partial sums.
// Each block: 64x64 pixel tile, 256 threads (8 wave32), 16 pixels/thread.
// Both images staged with halo into LDS (2 * 66*66 * 4B = 34.8 KB of 320 KB WGP LDS).
// Entire divergence computed in log2 domain; single *ln2 per thread at the end.
__global__ __launch_bounds__(256) void jsd_tile_kernel(const float* __restrict__ A,
                                                       const float* __restrict__ B,
                                                       float* __restrict__ partial) {
    __shared__ float ta[HS * HS];
    __shared__ float tb[HS * HS];
    __shared__ float red[16];

    const int tid = threadIdx.x;
    const int x0 = blockIdx.x * TILE - 1;   // halo origin
    const int y0 = blockIdx.y * TILE - 1;

    // Cooperative halo load with zero padding (matches 'SAME' zero-pad unfold).
    for (int i = tid; i < HS * HS; i += 256) {
        const int ly = i / HS;
        const int lx = i - ly * HS;
        const int gy = y0 + ly;
        const int gx = x0 + lx;
        float va = 0.0f, vb = 0.0f;
        if (gx >= 0 && gx < IMG_W && gy >= 0 && gy < IMG_H) {
            const int g = gy * IMG_W + gx;
            va = A[g];
            vb = B[g];
        }
        ta[i] = va;
        tb[i] = vb;
    }
    __syncthreads();

    const int tx  = tid & 63;   // column within tile (lanes -> consecutive LDS banks)
    const int ty0 = tid >> 6;   // 0..3; rows striped by 4

    float acc2 = 0.0f;          // log2-domain accumulator of (kl_pq + kl_qp)
                                // reference output = sum over pixels of kl = ln2 * acc2
    for (int k = 0; k < 16; ++k) {
        const int r = ty0 + 4 * k;                 // row within tile
        const int c = (r + 1) * HS + (tx + 1);     // center in halo coords

        float d2a[9], ga[9], d2b[9], gb[9];
        float Sa = 0.0f, Sb = 0.0f;
        const float ca = ta[c];
        const float cb = tb[c];

        int idx = 0;
#pragma unroll
        for (int dy = -1; dy <= 1; ++dy) {
#pragma unroll
            for (int dx = -1; dx <= 1; ++dx) {
                const float na = ta[c + dy * HS + dx];
                const float nb = tb[c + dy * HS + dx];
                float da = na - ca; da = da * da;      // diff^2 <= 1 (inputs in [0,1))
                float db = nb - cb; db = db * db;
                const float ea = hw_exp2(C2 * da);     // v_exp_f32, no extra mul
                const float eb = hw_exp2(C2 * db);
                d2a[idx] = da; ga[idx] = ea; Sa += ea;
                d2b[idx] = db; gb[idx] = eb; Sb += eb;
                ++idx;
            }
        }

        const float invSa = __builtin_amdgcn_rcpf(Sa);  // Sa in [~5.4, 9]: safe fast rcp
        const float invSb = __builtin_amdgcn_rcpf(Sb);
        // Fold the "+1" from log2(m) = log2(p+q) - 1 into the per-pixel constants:
        const float ka = 1.0f - hw_log2(Sa);            // v_log_f32, no extra mul
        const float kb = 1.0f - hw_log2(Sb);

        float kl2 = 0.0f;
#pragma unroll
        for (int i = 0; i < 9; ++i) {
            const float p   = ga[i] * invSa;
            const float q   = gb[i] * invSb;
            const float lp1 = __fmaf_rn(C2, d2a[i], ka); // log2(p) + 1
            const float lq1 = __fmaf_rn(C2, d2b[i], kb); // log2(q) + 1
            const float ls  = hw_log2(p + q);            // log2(2m); the only per-elem log
            kl2 = __fmaf_rn(p, lp1 - ls, kl2);
            kl2 = __fmaf_rn(q, lq1 - ls, kl2);
        }
        acc2 += kl2;
    }

    const float acc = acc2 * LN2;   // one base conversion per thread, not per log

    // Wave-level reduction using V_WMMA_F32_16X16X4_F32 (full EXEC here):
    // A(16x4) holds lane partials (lane m -> A[m][0], lane m+16 -> A[m][2], rest 0),
    // B = ones(4x16) => D[m][n] = acc_m + acc_{m+16}. Summing the 8 D VGPRs gives
    // lanes 0-15: sum(rows 0..7), lanes 16-31: sum(rows 8..15).
    v2f a; a.x = acc;  a.y = 0.0f;
    v2f b; b.x = 1.0f; b.y = 1.0f;
    v8f cz = {};
    v8f d = __builtin_amdgcn_wmma_f32_16x16x4_f32(false, a, false, b,
                                                  (short)0, cz, false, false);
    const float s = d[0] + d[1] + d[2] + d[3] + d[4] + d[5] + d[6] + d[7];

    const int lane = tid & 31;
    const int wave = tid >> 5;
    if ((lane & 15) == 0) red[wave * 2 + (lane >> 4)] = s;
    __syncthreads();

    if (tid == 0) {
        float t = 0.0f;
#pragma unroll
        for (int i = 0; i < 16; ++i) t += red[i];
        partial[blockIdx.y * gridDim.x + blockIdx.x] = t;
    }
}

// Stage-2: deterministic final reduction of block partials (1024 values).
__global__ __launch_bounds__(256) void jsd_reduce_kernel(const float* __restrict__ partial,
                                                         int n, float* __restrict__ out) {
    __shared__ float sm[256];
    float t = 0.0f;
    for (int i = threadIdx.x; i < n; i += 256) t += partial[i];
    sm[threadIdx.x] = t;
    __syncthreads();
    for (int s = 128; s > 0; s >>= 1) {
        if (threadIdx.x < s) sm[threadIdx.x] += sm[threadIdx.x + s];
        __syncthreads();
    }
    if (threadIdx.x == 0) out[0] = sm[0];
}

extern "C" void kernel_launch(void* const* d_in, const int* in_sizes, int n_in,
                              void* d_out, int out_size, void* d_ws, size_t ws_size,
                              hipStream_t stream) {
    const float* pred_real = (const float*)d_in[0];
    const float* pred_fake = (const float*)d_in[1];
    float* out     = (float*)d_out;
    float* partial = (float*)d_ws;   // (2048/64)^2 = 1024 floats = 4 KB

    dim3 grid(IMG_W / TILE, IMG_H / TILE);   // 32 x 32 blocks
    jsd_tile_kernel<<<grid, 256, 0, stream>>>(pred_real, pred_fake, partial);

    const int nblk = (IMG_W / TILE) * (IMG_H / TILE);
    jsd_reduce_kernel<<<1, 256, 0, stream>>>(partial, nblk, out);
}